// AdjointBilinearLayer_85048942395861
// MI455X (gfx1250) — compile-verified
//
#include <hip/hip_runtime.h>
#include <stdint.h>

#define ALG_DIM 248
#define TILE_B  64          // batch rows per workgroup tile
#define TS      66          // padded LDS tile stride in floats (conflict-free b64 lane access)
#define NWAVES  8
#define BLOCK   (NWAVES * 32)

__device__ __forceinline__ float readlane_f(float v, int lane) {
    return __uint_as_float(__builtin_amdgcn_readlane(__float_as_uint(v), lane));
}

__global__ __launch_bounds__(BLOCK) void
adjoint_bilinear_kernel(const float* __restrict__ xg,
                        const float* __restrict__ yg,
                        const int*   __restrict__ idx_i,
                        const int*   __restrict__ idx_j,
                        const int*   __restrict__ idx_k,
                        const float* __restrict__ coeff,
                        const float* __restrict__ alpha,
                        float* __restrict__ outg,
                        int batch, int nnz)
{
    // Transposed tiles: element (col c, row b) at [c*TS + b].  3 * 248*66*4B ~= 192 KB LDS.
    __shared__ float ldsX[ALG_DIM * TS];
    __shared__ float ldsY[ALG_DIM * TS];
    __shared__ float ldsO[ALG_DIM * TS];

    const int tid  = threadIdx.x;
    const int base = blockIdx.x * TILE_B;

    // ---- Stage x,y tiles into LDS (transposed) via CDNA5 async global->LDS gather copies.
    for (int n = tid; n < TILE_B * ALG_DIM; n += BLOCK) {
        const int b = n / ALG_DIM;        // row within tile, 0..63
        const int c = n - b * ALG_DIM;    // algebra index, 0..247
        if (base + b < batch) {
            const size_t goff = (size_t)(base + b) * ALG_DIM + c;
            uint32_t lx = (uint32_t)(uintptr_t)&ldsX[c * TS + b];
            uint64_t gx = (uint64_t)(uintptr_t)(xg + goff);
            asm volatile("global_load_async_to_lds_b32 %0, %1, off"
                         :: "v"(lx), "v"(gx) : "memory");
            uint32_t ly = (uint32_t)(uintptr_t)&ldsY[c * TS + b];
            uint64_t gy = (uint64_t)(uintptr_t)(yg + goff);
            asm volatile("global_load_async_to_lds_b32 %0, %1, off"
                         :: "v"(ly), "v"(gy) : "memory");
        }
    }

    // Zero the accumulator tile while the async copies are in flight.
    for (int n = tid; n < ALG_DIM * TS; n += BLOCK) ldsO[n] = 0.0f;

    const float a = alpha[0];            // scalar, folded into coeff below

    asm volatile("s_wait_asynccnt 0" ::: "memory");
    __syncthreads();

    // ---- Compute: 8 waves split the triple list. Lanes cooperatively load 32 triples
    // (4 coalesced b32 loads per chunk), then broadcast each triple to SGPRs with
    // v_readlane so the hot loop is pure SALU addressing + ds_load_b64 + ds_add_f32.
    const int wave = tid >> 5;
    const int lane = tid & 31;
    const int per  = (nnz + NWAVES - 1) / NWAVES;
    const int t0   = wave * per;
    const int t1   = (t0 + per < nnz) ? (t0 + per) : nnz;
    const int b2   = 2 * lane;

    for (int tc = t0; tc < t1; tc += 32) {
        const int  tl = tc + lane;
        const bool v  = (tl < t1);
        // Zero-coefficient padding for the tail: contributes exactly 0 to ldsO[0..].
        const int   vii = v ? idx_i[tl] : 0;
        const int   vjj = v ? idx_j[tl] : 0;
        const int   vkk = v ? idx_k[tl] : 0;
        const float vcf = (v ? coeff[tl] : 0.0f) * a;

        #pragma unroll
        for (int n = 0; n < 32; ++n) {
            const int   ii = __builtin_amdgcn_readlane(vii, n);
            const int   jj = __builtin_amdgcn_readlane(vjj, n);
            const int   kk = __builtin_amdgcn_readlane(vkk, n);
            const float cf = readlane_f(vcf, n);
            const float2 xv = *(const float2*)&ldsX[ii * TS + b2];
            const float2 yv = *(const float2*)&ldsY[jj * TS + b2];
            float* o = &ldsO[kk * TS + b2];
            atomicAdd(o,     cf * xv.x * yv.x);   // ds_add_f32
            atomicAdd(o + 1, cf * xv.y * yv.y);   // ds_add_f32
        }
    }

    __syncthreads();

    // ---- Write out: accumulator already holds alpha-scaled values, so this is a
    // pure LDS -> global copy via the CDNA5 async store path (no VGPR round-trip).
    for (int n = tid; n < TILE_B * ALG_DIM; n += BLOCK) {
        const int b = n / ALG_DIM;
        const int k = n - b * ALG_DIM;
        if (base + b < batch) {
            uint32_t lo = (uint32_t)(uintptr_t)&ldsO[k * TS + b];
            uint64_t go = (uint64_t)(uintptr_t)(outg + (size_t)(base + b) * ALG_DIM + k);
            asm volatile("global_store_async_from_lds_b32 %0, %1, off"
                         :: "v"(go), "v"(lo) : "memory");
        }
    }
    asm volatile("s_wait_asynccnt 0" ::: "memory");
}

extern "C" void kernel_launch(void* const* d_in, const int* in_sizes, int n_in,
                              void* d_out, int out_size, void* d_ws, size_t ws_size,
                              hipStream_t stream) {
    const float* x  = (const float*)d_in[0];
    const float* y  = (const float*)d_in[1];
    const int*   ii = (const int*)d_in[2];
    const int*   jj = (const int*)d_in[3];
    const int*   kk = (const int*)d_in[4];
    const float* cf = (const float*)d_in[5];
    const float* al = (const float*)d_in[6];
    float* out = (float*)d_out;

    const int nnz   = in_sizes[2];
    const int batch = in_sizes[0] / ALG_DIM;
    const int grid  = (batch + TILE_B - 1) / TILE_B;

    hipLaunchKernelGGL(adjoint_bilinear_kernel, dim3(grid), dim3(BLOCK), 0, stream,
                       x, y, ii, jj, kk, cf, al, out, batch, nnz);
}